// RMILoss_35948876268211
// MI455X (gfx1250) — compile-verified
//
#include <hip/hip_runtime.h>
#include <math.h>

#define NCLS 21
#define NB 4
#define H 512
#define W 512
#define HW (H*W)
#define PH 171
#define PHW (PH*PH)            // 29241
#define NHD 169
#define M_SAMPLES (NHD*NHD)    // 28561
#define DDIM 9
#define NPAIR (NB*NCLS)        // 84
#define POS_ALPHA 5e-4f
#define CLIP_MIN 1e-6f
#define BCE_BLOCKS 4096

typedef __attribute__((ext_vector_type(2))) float v2f;
typedef __attribute__((ext_vector_type(8))) float v8f;

__device__ __forceinline__ float softplusf(float x) {
    // log(1+exp(x)), branchless-stable
    return fmaxf(x, 0.f) + log1pf(__expf(-fabsf(x)));
}

// ---------------- Kernel 1: BCE-with-logits, per-block partial sums ----------------
__global__ void __launch_bounds__(256)
bce_kernel(const float* __restrict__ logits, const int* __restrict__ labels,
           float* __restrict__ partial, float* __restrict__ pcount) {
    int idx = blockIdx.x * 256 + threadIdx.x;       // exactly NB*HW threads
    int n   = idx / HW;
    int hw  = idx % HW;
    int lab = labels[idx];
    float mask = (lab < NCLS) ? 1.f : 0.f;
    int  labm  = (lab < NCLS) ? lab : 0;
    const float* lg = logits + (size_t)n * NCLS * HW + hw;
    float s = 0.f;
    #pragma unroll
    for (int c = 0; c < NCLS; ++c) {
        float x = lg[(size_t)c * HW];
        float y = (c == labm) ? mask : 0.f;
        // y*sp(-x) + (1-y)*sp(x) == sp(x) - y*x
        s += softplusf(x) - y * x;
    }
    float local = s * mask;

    __shared__ float sl[256], sc[256];
    sl[threadIdx.x] = local;
    sc[threadIdx.x] = mask;
    __syncthreads();
    for (int st = 128; st > 0; st >>= 1) {
        if (threadIdx.x < st) {
            sl[threadIdx.x] += sl[threadIdx.x + st];
            sc[threadIdx.x] += sc[threadIdx.x + st];
        }
        __syncthreads();
    }
    if (threadIdx.x == 0) {
        partial[blockIdx.x] = sl[0];
        pcount[blockIdx.x]  = sc[0];
    }
}

// --------- Kernel 2: fused sigmoid*mask+clip and onehot, 3x3/s3/p1 max-pool --------
__global__ void __launch_bounds__(256)
pool_kernel(const float* __restrict__ logits, const int* __restrict__ labels,
            float* __restrict__ Pp, float* __restrict__ Lp) {
    int idx = blockIdx.x * 256 + threadIdx.x;
    if (idx >= NPAIR * PHW) return;
    int b   = idx / PHW;
    int rem = idx % PHW;
    int ph  = rem / PH, pw = rem % PH;
    int n   = b / NCLS, c = b % NCLS;
    const float* lg = logits + (size_t)(n * NCLS + c) * HW;
    const int*   lb = labels + (size_t)n * HW;
    float mp = -INFINITY, ml = -INFINITY;
    #pragma unroll
    for (int dy = 0; dy < 3; ++dy) {
        int hh = 3 * ph - 1 + dy;
        if (hh < 0 || hh >= H) continue;
        #pragma unroll
        for (int dx = 0; dx < 3; ++dx) {
            int ww = 3 * pw - 1 + dx;
            if (ww < 0 || ww >= W) continue;
            int lab = lb[hh * W + ww];
            float mask = (lab < NCLS) ? 1.f : 0.f;
            float x = lg[hh * W + ww];
            float p = mask / (1.f + __expf(-x)) + CLIP_MIN;   // sigmoid*mask + clip
            float l = (lab == c) ? mask : 0.f;
            mp = fmaxf(mp, p);
            ml = fmaxf(ml, l);
        }
    }
    Pp[idx] = mp;
    Lp[idx] = ml;
}

// --- Kernel 3: per-(n,c) 9x9 Grams via V_WMMA_F32_16X16X4_F32 + 9x9 linear algebra ---
// Row-major sample enumeration: waves own interleaved rows, columns walk linearly,
// so the hot loop has no division and no divergent branches (EXEC stays all-1s).
// Pad-rows 9..15 of the 16x16 tile load in-workspace garbage; it only reaches
// rows/cols 9..15 of the accumulator, which the reduction never reads.
__global__ void __launch_bounds__(256)
gram_kernel(const float* __restrict__ Pp, const float* __restrict__ Lp,
            float* __restrict__ rmi) {
    const int b = blockIdx.x;                 // 0..83
    const float* Pb = Pp + (size_t)b * PHW;
    const float* Lb = Lp + (size_t)b * PHW;
    const int lane = threadIdx.x & 31;
    const int wv   = threadIdx.x >> 5;        // wave 0..7
    const int r    = lane & 15;               // Gram row d (0..8 valid, 9..15 pad)
    const int hi   = lane >> 4;
    const int dy   = r / 3, dx = r % 3;       // shift offsets (garbage-safe for r>=9)
    const int lcol = dx + 2 * hi;             // per-lane column base within a row
    const int   toff  = (hi == 0) ? 168 : 0;  // tail-column offset (loop-invariant)
    const float hmask = (hi == 0) ? 1.f : 0.f;

    v8f accLL = {0.f,0.f,0.f,0.f,0.f,0.f,0.f,0.f};
    v8f accPP = accLL;
    v8f accLP = accLL;
    float sumL = 0.f, sumP = 0.f;

    for (int i = wv; i < NHD; i += 8) {
        const float* lrow = Lb + (i + dy) * PH + lcol;
        const float* prow = Pb + (i + dy) * PH + lcol;
        // 42 full K=4 chunks: columns 0..167, fully unguarded
        for (int j0 = 0; j0 < NHD - 1; j0 += 4) {
            float l0 = lrow[j0];
            float l1 = lrow[j0 + 1];
            float p0 = prow[j0];
            float p1 = prow[j0 + 1];
            sumL += l0 + l1;
            sumP += p0 + p1;
            v2f a_l = {l0, l1};
            v2f a_p = {p0, p1};
            accLL = __builtin_amdgcn_wmma_f32_16x16x4_f32(false, a_l, false, a_l, (short)0, accLL, false, false);
            accPP = __builtin_amdgcn_wmma_f32_16x16x4_f32(false, a_p, false, a_p, (short)0, accPP, false, false);
            accLP = __builtin_amdgcn_wmma_f32_16x16x4_f32(false, a_l, false, a_p, (short)0, accLP, false, false);
        }
        // tail chunk: only column 168 is valid (k=0 of the hi==0 half)
        {
            float lt = hmask * lrow[toff];
            float pt = hmask * prow[toff];
            sumL += lt;
            sumP += pt;
            v2f a_l = {lt, 0.f};
            v2f a_p = {pt, 0.f};
            accLL = __builtin_amdgcn_wmma_f32_16x16x4_f32(false, a_l, false, a_l, (short)0, accLL, false, false);
            accPP = __builtin_amdgcn_wmma_f32_16x16x4_f32(false, a_p, false, a_p, (short)0, accPP, false, false);
            accLP = __builtin_amdgcn_wmma_f32_16x16x4_f32(false, a_l, false, a_p, (short)0, accLP, false, false);
        }
    }

    // --- cross-wave reduction of the three 16x16 accumulators through LDS ---
    __shared__ float sAcc[3][8][256];
    #pragma unroll
    for (int v = 0; v < 8; ++v) {
        sAcc[0][wv][v * 32 + lane] = accLL[v];
        sAcc[1][wv][v * 32 + lane] = accPP[v];
        sAcc[2][wv][v * 32 + lane] = accLP[v];
    }
    __shared__ float sSL[256], sSP[256];
    sSL[threadIdx.x] = sumL;
    sSP[threadIdx.x] = sumP;
    __syncthreads();

    __shared__ float g[3][DDIM][DDIM];
    __shared__ float rsL[DDIM], rsP[DDIM];
    int t = threadIdx.x;
    if (t < 3 * 81) {
        int mat = t / 81, rc = t % 81, row = rc / 9, col = rc % 9;
        // C/D layout: VGPR v holds rows v (lanes 0-15, N=lane) and v+8 (lanes 16-31)
        int v  = row & 7;
        int ln = col + 16 * (row >> 3);
        float s = 0.f;
        #pragma unroll
        for (int wq = 0; wq < 8; ++wq) s += sAcc[mat][wq][v * 32 + ln];
        g[mat][row][col] = s;
    }
    if (t >= 243 && t < 243 + DDIM) {
        int d = t - 243;
        float s = 0.f;
        #pragma unroll
        for (int wq = 0; wq < 8; ++wq) s += sSL[wq * 32 + d] + sSL[wq * 32 + d + 16];
        rsL[d] = s;
    }
    if (t >= 208 && t < 208 + DDIM) {
        int d = t - 208;
        float s = 0.f;
        #pragma unroll
        for (int wq = 0; wq < 8; ++wq) s += sSP[wq * 32 + d] + sSP[wq * 32 + d + 16];
        rsP[d] = s;
    }
    __syncthreads();

    if (t == 0) {
        const float Mf = (float)M_SAMPLES;
        float mL[DDIM], mP[DDIM];
        for (int d = 0; d < DDIM; ++d) { mL[d] = rsL[d] / Mf; mP[d] = rsP[d] / Mf; }
        float laCov[DDIM][DDIM], prCov[DDIM][DDIM], lpCov[DDIM][DDIM];
        for (int i = 0; i < DDIM; ++i)
            for (int j = 0; j < DDIM; ++j) {
                laCov[i][j] = g[0][i][j] - Mf * mL[i] * mL[j];
                prCov[i][j] = g[1][i][j] - Mf * mP[i] * mP[j] + ((i == j) ? POS_ALPHA : 0.f);
                lpCov[i][j] = g[2][i][j] - Mf * mL[i] * mP[j];
            }
        // invert SPD prCov via Gauss-Jordan (diagonal regularized, no pivoting needed)
        float inv[DDIM][DDIM];
        for (int i = 0; i < DDIM; ++i)
            for (int j = 0; j < DDIM; ++j) inv[i][j] = (i == j) ? 1.f : 0.f;
        for (int k = 0; k < DDIM; ++k) {
            float ip = 1.f / prCov[k][k];
            for (int j = 0; j < DDIM; ++j) { prCov[k][j] *= ip; inv[k][j] *= ip; }
            for (int i = 0; i < DDIM; ++i) {
                if (i == k) continue;
                float f = prCov[i][k];
                for (int j = 0; j < DDIM; ++j) {
                    prCov[i][j] -= f * prCov[k][j];
                    inv[i][j]   -= f * inv[k][j];
                }
            }
        }
        // T = lpCov * inv ; B = laCov - T*lpCov^T + alpha*I
        float T[DDIM][DDIM];
        for (int i = 0; i < DDIM; ++i)
            for (int j = 0; j < DDIM; ++j) {
                float s = 0.f;
                for (int k = 0; k < DDIM; ++k) s += lpCov[i][k] * inv[k][j];
                T[i][j] = s;
            }
        float Bm[DDIM][DDIM];
        for (int i = 0; i < DDIM; ++i)
            for (int j = 0; j < DDIM; ++j) {
                float s = 0.f;
                for (int k = 0; k < DDIM; ++k) s += T[i][k] * lpCov[j][k];
                Bm[i][j] = laCov[i][j] - s + ((i == j) ? POS_ALPHA : 0.f);
            }
        // Cholesky; accumulate sum(log(diag + 1e-8))
        float Lc[DDIM][DDIM];
        float acc = 0.f;
        for (int i = 0; i < DDIM; ++i) {
            for (int j = 0; j <= i; ++j) {
                float s = Bm[i][j];
                for (int k = 0; k < j; ++k) s -= Lc[i][k] * Lc[j][k];
                if (i == j) {
                    float dg = sqrtf(s);
                    Lc[i][j] = dg;
                    acc += logf(dg + 1e-8f);
                } else {
                    Lc[i][j] = s / Lc[j][j];
                }
            }
        }
        rmi[b] = acc;
    }
}

// ---------------- Kernel 4: deterministic final combine ----------------
__global__ void __launch_bounds__(256)
finalize_kernel(const float* __restrict__ partial, const float* __restrict__ pcount,
                const float* __restrict__ rmi, float* __restrict__ out, int nblocks) {
    __shared__ float sb[256], sc[256];
    float a = 0.f, c2 = 0.f;
    for (int i = threadIdx.x; i < nblocks; i += 256) { a += partial[i]; c2 += pcount[i]; }
    sb[threadIdx.x] = a;
    sc[threadIdx.x] = c2;
    __syncthreads();
    for (int st = 128; st > 0; st >>= 1) {
        if (threadIdx.x < st) {
            sb[threadIdx.x] += sb[threadIdx.x + st];
            sc[threadIdx.x] += sc[threadIdx.x + st];
        }
        __syncthreads();
    }
    if (threadIdx.x == 0) {
        float bce = sb[0] / (sc[0] + 1.f);
        float rtot = 0.f;
        for (int c = 0; c < NCLS; ++c) {
            float s = 0.f;
            for (int n = 0; n < NB; ++n) s += rmi[n * NCLS + c];
            rtot += (s / (float)NB) / (float)DDIM;
        }
        out[0] = 0.5f * bce + 0.5f * rtot;
    }
}

extern "C" void kernel_launch(void* const* d_in, const int* in_sizes, int n_in,
                              void* d_out, int out_size, void* d_ws, size_t ws_size,
                              hipStream_t stream) {
    const float* logits = (const float*)d_in[0];
    const int*   labels = (const int*)d_in[1];
    float* ws      = (float*)d_ws;
    float* Pp      = ws;                                  // 84*29241
    float* Lp      = Pp + (size_t)NPAIR * PHW;            // 84*29241
    float* pad     = Lp + (size_t)NPAIR * PHW;            // 1024 (absorbs pad-row reads)
    float* partial = pad + 1024;                          // 4096
    float* pcount  = partial + BCE_BLOCKS;                // 4096
    float* rmi     = pcount + BCE_BLOCKS;                 // 84

    bce_kernel<<<BCE_BLOCKS, 256, 0, stream>>>(logits, labels, partial, pcount);
    int pool_total = NPAIR * PHW;
    pool_kernel<<<(pool_total + 255) / 256, 256, 0, stream>>>(logits, labels, Pp, Lp);
    gram_kernel<<<NPAIR, 256, 0, stream>>>(Pp, Lp, rmi);
    finalize_kernel<<<1, 256, 0, stream>>>(partial, pcount, rmi, (float*)d_out, BCE_BLOCKS);
}